// MyModel_61933428411029
// MI455X (gfx1250) — compile-verified
//
#include <hip/hip_runtime.h>

typedef float v2f __attribute__((ext_vector_type(2)));
typedef float v4f __attribute__((ext_vector_type(4)));
typedef float v8f __attribute__((ext_vector_type(8)));

#define B_     128
#define L_     16384
#define C_     128
#define NT_    128
#define OUTL_  16513            // L + 2*ntaps - (ntaps-1)
#define TILE_  256              // outputs per wave-tile (16 m x 16 n)
#define NTILES_ 65              // ceil(16513/256)
#define WAVES_ 8
#define NCHUNK_ 36              // ceil((128+15)/4) chunks of K=4 taps
#define BLKWIN_ (WAVES_ * TILE_ + 256)   // 2304: [J0-128, J0+2048+128)

// ---------------- Kernel 1: S[b] = sum_l x[b,l] ----------------
__global__ void k_rowsum(const float* __restrict__ x, float* __restrict__ S) {
    __shared__ float sdata[256];
    const int b = blockIdx.x;
    const v4f* xb4 = (const v4f*)(x + (size_t)b * L_);   // rows are 16B aligned
    float acc = 0.f;
    for (int i = threadIdx.x; i < L_ / 4; i += 256) {
        v4f v = xb4[i];
        acc += (v.x + v.y) + (v.z + v.w);
    }
    sdata[threadIdx.x] = acc;
    __syncthreads();
    for (int s = 128; s > 0; s >>= 1) {
        if (threadIdx.x < s) sdata[threadIdx.x] += sdata[threadIdx.x + s];
        __syncthreads();
    }
    if (threadIdx.x == 0) S[b] = sdata[0];
}

// ---------------- Kernel 2: filt[b,t] = fc_b[t] + sum_c fc_w[t,c]*pooled[b,c]
// pooled[b,c] = conv_b[c] + (w0*(S-xlast) + w1*S + w2*(S-x0)) / L  (mean-of-conv algebra)
__global__ void k_filt(const float* __restrict__ x,
                       const float* __restrict__ cw,  // (C,1,3)
                       const float* __restrict__ cb,  // (C,)
                       const float* __restrict__ fw,  // (ntaps, C)
                       const float* __restrict__ fb,  // (ntaps,)
                       const float* __restrict__ S,
                       float* __restrict__ filt) {
    __shared__ float pooled[C_];
    const int b = blockIdx.x;
    const int t = threadIdx.x;          // 128 threads
    const float Sb = S[b];
    const float x0 = x[(size_t)b * L_];
    const float xl = x[(size_t)b * L_ + (L_ - 1)];
    {
        const int c = t;
        const float w0 = cw[c * 3 + 0], w1 = cw[c * 3 + 1], w2 = cw[c * 3 + 2];
        pooled[c] = cb[c] + (w0 * (Sb - xl) + w1 * Sb + w2 * (Sb - x0)) * (1.0f / (float)L_);
    }
    __syncthreads();
    float acc = fb[t];
    #pragma unroll 8
    for (int c = 0; c < C_; ++c) acc += fw[t * C_ + c] * pooled[c];
    filt[b * NT_ + t] = acc;
}

// ---------------- Kernel 3: per-batch 128-tap FIR via f32 WMMA ----------------
// y[b,j] = sum_{t=0..127} filt[b,t] * x[b, j + t - 128],  j in [0, OUTL)
// Wave-tile of 256 outputs: D[m,n] = y[j0 + m + 16n].
// Chunk c = 4*ci:  A[m,k] = filt_pad[c + k - m],  B[k,n] = x_pad[j0 - 128 + 16n + c + k]
// Two independent accumulator chains (even/odd ci) keep WMMAs issuing back-to-back.
__global__ void k_fir_wmma(const float* __restrict__ x,
                           const float* __restrict__ filt,
                           float* __restrict__ out) {
    __shared__ float lds_filt[160];        // 16 zero-pad | 128 taps | 16 zero-pad
    __shared__ float lds_x[BLKWIN_];       // shared window [J0-128, J0+2048+128)
    const int b    = blockIdx.y;
    const int wave = threadIdx.x >> 5;
    const int lane = threadIdx.x & 31;
    const int tile = blockIdx.x * WAVES_ + wave;
    const int J0   = blockIdx.x * (WAVES_ * TILE_);

    // padded filter to LDS (same b for whole block)
    for (int i = threadIdx.x; i < 160; i += blockDim.x) {
        const int t = i - 16;
        lds_filt[i] = (t >= 0 && t < NT_) ? filt[b * NT_ + t] : 0.f;
    }
    // block-shared x window, coalesced: lds_x[s] = x[b, J0 - 128 + s]
    const float* xb = x + (size_t)b * L_;
    for (int i = threadIdx.x; i < BLKWIN_; i += blockDim.x) {
        const int g = J0 - 128 + i;
        lds_x[i] = (g >= 0 && g < L_) ? xb[g] : 0.f;
    }
    __syncthreads();

    if (tile < NTILES_) {                  // wave-uniform: EXEC all-ones for WMMA
        const int mn   = lane & 15;        // row m for A, column n for B/D
        const int half = lane >> 4;
        const int kk   = half * 2;         // this lane holds K = {kk, kk+1}
        const float* xw = &lds_x[wave * TILE_];   // xw[s] = x_pad[j0 - 128 + s]
        v8f acc0 = {0.f, 0.f, 0.f, 0.f, 0.f, 0.f, 0.f, 0.f};
        v8f acc1 = {0.f, 0.f, 0.f, 0.f, 0.f, 0.f, 0.f, 0.f};
        #pragma unroll
        for (int ci = 0; ci < NCHUNK_; ci += 2) {
            // ---- even chunk -> acc0 ----
            {
                const int c  = ci * 4;
                const int aq = c + kk - mn + 16;     // in [1, 159]
                v2f a; a.x = lds_filt[aq]; a.y = lds_filt[aq + 1];
                const int bq = mn * 16 + c + kk;     // even -> 8B aligned
                const v2f bv = *(const v2f*)(xw + bq);
                acc0 = __builtin_amdgcn_wmma_f32_16x16x4_f32(
                    false, a, false, bv, (short)0, acc0, false, false);
            }
            // ---- odd chunk -> acc1 ----
            {
                const int c  = (ci + 1) * 4;
                const int aq = c + kk - mn + 16;
                v2f a; a.x = lds_filt[aq]; a.y = lds_filt[aq + 1];
                const int bq = mn * 16 + c + kk;
                const v2f bv = *(const v2f*)(xw + bq);
                acc1 = __builtin_amdgcn_wmma_f32_16x16x4_f32(
                    false, a, false, bv, (short)0, acc1, false, false);
            }
        }
        const v8f acc = acc0 + acc1;

        // D layout: lane -> n = mn, vgpr r -> m = r + 8*half => j = j0 + 16n + 8*half + r
        const int j0    = tile * TILE_;
        const int jbase = j0 + 16 * mn + 8 * half;
        float* ob = out + (size_t)b * OUTL_;
        if (tile < NTILES_ - 1) {
            // full tile: unguarded contiguous stores -> backend merges to B128
            #pragma unroll
            for (int r = 0; r < 8; ++r) ob[jbase + r] = acc[r];
        } else {
            #pragma unroll
            for (int r = 0; r < 8; ++r) {
                const int j = jbase + r;
                if (j < OUTL_) ob[j] = acc[r];
            }
        }
    }
}

extern "C" void kernel_launch(void* const* d_in, const int* in_sizes, int n_in,
                              void* d_out, int out_size, void* d_ws, size_t ws_size,
                              hipStream_t stream) {
    const float* x      = (const float*)d_in[0];
    const float* conv_w = (const float*)d_in[1];
    const float* conv_b = (const float*)d_in[2];
    const float* fc_w   = (const float*)d_in[3];
    const float* fc_b   = (const float*)d_in[4];
    (void)in_sizes; (void)n_in; (void)out_size; (void)ws_size;

    float* wsf  = (float*)d_ws;
    float* S    = wsf;            // 128 floats
    float* filt = wsf + 128;      // 128*128 floats
    float* out  = (float*)d_out;  // 128 * 16513 floats

    k_rowsum<<<dim3(B_), dim3(256), 0, stream>>>(x, S);
    k_filt  <<<dim3(B_), dim3(C_), 0, stream>>>(x, conv_w, conv_b, fc_w, fc_b, S, filt);
    dim3 grid((NTILES_ + WAVES_ - 1) / WAVES_, B_);
    k_fir_wmma<<<grid, dim3(WAVES_ * 32), 0, stream>>>(x, filt, out);
}